// VectorQuantizerEMA_23837068492941
// MI455X (gfx1250) — compile-verified
//
#include <hip/hip_runtime.h>
#include <math.h>

// ---------------------------------------------------------------------------
// VQ-VAE EMA vector quantizer for MI455X (gfx1250, wave32, WMMA f32).
// N = 65536 rows, D = 64, K = 1024 codes.
// Main GEMM: distances via v_wmma_f32_16x16x4_f32 with LDS-staged,
// double-buffered codebook tiles so loads overlap the WMMA chain.
// ---------------------------------------------------------------------------

#define EMBED_DIM   64
#define NUM_EMB     1024
#define N_ROWS      65536
#define ROWS_PER_WAVE   16
#define WAVES_PER_BLOCK 8
#define ROWS_PER_BLOCK  (ROWS_PER_WAVE * WAVES_PER_BLOCK)   // 128
#define N_BLOCKS        (N_ROWS / ROWS_PER_BLOCK)           // 512
#define N_TILES         (NUM_EMB / 16)                      // 64
#define BPAD            68   // padded B-tile row stride (bank-conflict-free)

// Output layout (float32, flat concat of the reference tuple)
#define O_LOSS   ((size_t)0)
#define O_QUANT  ((size_t)1)
#define O_PERP   ((size_t)4194305)      // 1 + 65536*64
#define O_ENC    ((size_t)4194306)
#define O_EMBW   ((size_t)71303170)     // O_ENC + 65536*1024
#define O_EMAW   ((size_t)71368706)
#define O_NCS    ((size_t)71434242)

typedef __attribute__((ext_vector_type(2))) float v2f;
typedef __attribute__((ext_vector_type(8))) float v8f;

// ---------------------------------------------------------------------------
// Grid-stride zero fill (encodings region + workspace accumulators).
// ---------------------------------------------------------------------------
__global__ void zero_f32(float* __restrict__ p, size_t n) {
    size_t i = (size_t)blockIdx.x * blockDim.x + threadIdx.x;
    size_t stride = (size_t)gridDim.x * blockDim.x;
    for (; i < n; i += stride) p[i] = 0.0f;
}

// ---------------------------------------------------------------------------
// Per-code squared L2 norms: wnorm[k] = sum_d emb[k][d]^2
// ---------------------------------------------------------------------------
__global__ void wnorm_kernel(const float* __restrict__ emb,
                             float* __restrict__ wnorm) {
    int k = blockIdx.x * blockDim.x + threadIdx.x;
    if (k < NUM_EMB) {
        const float4* row = (const float4*)(emb + (size_t)k * EMBED_DIM);
        float s = 0.0f;
        #pragma unroll
        for (int i = 0; i < EMBED_DIM / 4; ++i) {
            float4 v = row[i];
            s += v.x * v.x + v.y * v.y + v.z * v.z + v.w * v.w;
        }
        wnorm[k] = s;
    }
}

// ---------------------------------------------------------------------------
// Main kernel: WMMA f32 distance GEMM + argmin + gather/scatter.
// Each wave owns 16 rows (A in registers). The block cooperatively stages
// each 16-code B tile in LDS (double-buffered; refill overlaps WMMAs).
// Score = ||w||^2 - 2 * <x, w>   (||x||^2 is constant per row for argmin)
// ---------------------------------------------------------------------------
__global__ __launch_bounds__(256) void vq_main(
    const float* __restrict__ flat,    // [N, 64]
    const float* __restrict__ emb,     // [1024, 64]
    const float* __restrict__ wnorm,   // [1024]
    float* __restrict__ out_quant,     // -> d_out + O_QUANT
    float* __restrict__ out_enc,       // -> d_out + O_ENC (pre-zeroed)
    float* __restrict__ ws_count,      // [1024]   (pre-zeroed)
    float* __restrict__ ws_dw,         // [1024*64](pre-zeroed)
    float* __restrict__ ws_loss)       // [1]      (pre-zeroed)
{
    __shared__ float s_b[2][16][BPAD];               // double-buffered B tile
    __shared__ float s_val[WAVES_PER_BLOCK][16][16];
    __shared__ int   s_idx[WAVES_PER_BLOCK][16][16];
    __shared__ int   s_code[ROWS_PER_BLOCK];
    __shared__ float s_red[256];

    const int tid  = threadIdx.x;
    const int wave = tid >> 5;
    const int lane = tid & 31;
    const int half = lane >> 4;     // 0: lanes 0-15, 1: lanes 16-31
    const int lm   = lane & 15;

    // B-tile fill assignment: thread -> (row fr, 4-float column fc).
    // Tile nt = 16 consecutive codebook rows = 1024 contiguous floats.
    const int fr = tid >> 4;          // 0..15
    const int fc = (tid & 15) * 4;    // 0..60

    const int rowbase = blockIdx.x * ROWS_PER_BLOCK + wave * ROWS_PER_WAVE;

    // ---- Preload A fragments (16x4 f32 per WMMA, 16 k-steps cover D=64).
    // A layout: lanes 0-15 hold M=0..15 with K={0,1}; lanes 16-31 K={2,3}.
    v2f a[16];
    {
        const float* arow = flat + (size_t)(rowbase + lm) * EMBED_DIM + 2 * half;
        #pragma unroll
        for (int s = 0; s < 16; ++s)
            a[s] = *(const v2f*)(arow + 4 * s);
    }

    // ---- Stage B tile 0 into LDS buffer 0.
    {
        float4 v = *(const float4*)(emb + (size_t)fr * EMBED_DIM + fc);
        *(float4*)(&s_b[0][fr][fc]) = v;
    }
    __syncthreads();

    float minval[8];
    int   minidx[8];
    #pragma unroll
    for (int r = 0; r < 8; ++r) { minval[r] = 3.4e38f; minidx[r] = 0; }

    float4 stage = make_float4(0.f, 0.f, 0.f, 0.f);

    for (int nt = 0; nt < N_TILES; ++nt) {
        const int buf = nt & 1;

        // Issue next tile's global load first (latency hidden by WMMAs).
        if (nt + 1 < N_TILES) {
            stage = *(const float4*)(emb + (size_t)(nt + 1) * 1024
                                         + (size_t)fr * EMBED_DIM + fc);
            if (nt + 2 < N_TILES)
                __builtin_prefetch(emb + (size_t)(nt + 2) * 1024 + tid * 4, 0, 3);
        }

        // B fragments from LDS (batched so ds_loads can clause together).
        // B[k][n] = emb[n][k]: lane holds n=lm, K={0,1}/{2,3} per step.
        v2f b[16];
        #pragma unroll
        for (int s = 0; s < 16; ++s)
            b[s] = *(const v2f*)(&s_b[buf][lm][4 * s + 2 * half]);

        v8f acc = {};
        #pragma unroll
        for (int s = 0; s < 16; ++s)
            acc = __builtin_amdgcn_wmma_f32_16x16x4_f32(
                /*neg_a=*/false, a[s], /*neg_b=*/false, b[s],
                /*c_mod=*/(short)0, acc, /*reuse_a=*/false, /*reuse_b=*/false);

        // C/D layout: VGPR r, lanes 0-15 -> M=r, N=lane; lanes 16-31 -> M=r+8.
        const int   code = nt * 16 + lm;
        const float wn   = wnorm[code];
        #pragma unroll
        for (int r = 0; r < 8; ++r) {
            float v = wn - 2.0f * acc[r];
            if (v < minval[r]) { minval[r] = v; minidx[r] = code; }
        }

        // Commit next tile to the other LDS buffer, then sync.
        if (nt + 1 < N_TILES)
            *(float4*)(&s_b[buf ^ 1][fr][fc]) = stage;
        __syncthreads();
    }

    // ---- Cross-lane argmin reduction via LDS (rows x 16 candidate lanes).
    #pragma unroll
    for (int r = 0; r < 8; ++r) {
        int rowl = r + 8 * half;
        s_val[wave][rowl][lm] = minval[r];
        s_idx[wave][rowl][lm] = minidx[r];
    }
    __syncthreads();

    if (tid < ROWS_PER_BLOCK) {
        int w = tid >> 4, rowl = tid & 15;
        float bv = s_val[w][rowl][0];
        int   bk = s_idx[w][rowl][0];
        #pragma unroll
        for (int j = 1; j < 16; ++j) {
            float v = s_val[w][rowl][j];
            int   k = s_idx[w][rowl][j];
            // jnp.argmin picks the first (lowest) index among equal minima.
            if (v < bv || (v == bv && k < bk)) { bv = v; bk = k; }
        }
        int rowg = blockIdx.x * ROWS_PER_BLOCK + tid;
        s_code[tid] = bk;
        atomicAdd(&ws_count[bk], 1.0f);
        out_enc[(size_t)rowg * NUM_EMB + bk] = 1.0f;   // one-hot
    }
    __syncthreads();

    // ---- Cooperative gather of quantized vectors, loss, and dw scatter.
    float lsum = 0.0f;
    const int base_elem = blockIdx.x * ROWS_PER_BLOCK * EMBED_DIM;
    for (int e = tid; e < ROWS_PER_BLOCK * EMBED_DIM; e += 256) {
        int rowl = e >> 6, d = e & 63;
        int k = s_code[rowl];
        float xv = flat[(size_t)base_elem + e];
        float q  = emb[(size_t)k * EMBED_DIM + d];
        out_quant[(size_t)base_elem + e] = q;          // straight-through == q
        float diff = q - xv;
        lsum += diff * diff;
        atomicAdd(&ws_dw[(size_t)k * EMBED_DIM + d], xv);  // dw.T accumulation
    }

    s_red[tid] = lsum;
    __syncthreads();
    for (int off = 128; off > 0; off >>= 1) {
        if (tid < off) s_red[tid] += s_red[tid + off];
        __syncthreads();
    }
    if (tid == 0) atomicAdd(ws_loss, s_red[0]);
}

// ---------------------------------------------------------------------------
// Finalize: EMA updates, renormalized codebook, perplexity, loss scalar.
// Single workgroup of 1024 threads (one per code).
// ---------------------------------------------------------------------------
__global__ __launch_bounds__(1024) void vq_finalize(
    const float* __restrict__ ema_w,     // [1024*64]
    const float* __restrict__ ema_cs,    // [1024]
    const float* __restrict__ ws_count,  // [1024]
    const float* __restrict__ ws_dw,     // [1024*64]
    const float* __restrict__ ws_loss,   // [1]
    float* __restrict__ out)             // full d_out
{
    __shared__ float s_n[1024];
    __shared__ float s_e[1024];

    const int t = threadIdx.x;
    const float cnt = ws_count[t];
    const float ncs = ema_cs[t] * 0.99f + 0.01f * cnt;   // DECAY = 0.99
    out[O_NCS + t] = ncs;

    const float p = cnt * (1.0f / (float)N_ROWS);
    s_n[t] = ncs;
    s_e[t] = p * logf(p + 1e-10f);
    __syncthreads();
    for (int off = 512; off > 0; off >>= 1) {
        if (t < off) { s_n[t] += s_n[t + off]; s_e[t] += s_e[t + off]; }
        __syncthreads();
    }
    const float nsum = s_n[0];
    const float ent  = s_e[0];

    const float updated = (ncs + 1e-5f) / (nsum + (float)NUM_EMB * 1e-5f) * nsum;
    const float inv_upd = 1.0f / updated;

    for (int d = 0; d < EMBED_DIM; ++d) {
        size_t i = (size_t)t * EMBED_DIM + d;
        float nw = ema_w[i] * 0.99f + 0.01f * ws_dw[i];
        out[O_EMAW + i] = nw;           // new_ema_w
        out[O_EMBW + i] = nw * inv_upd; // new_emb_weights
    }

    if (t == 0) {
        out[O_LOSS] = 0.25f * ws_loss[0] * (1.0f / ((float)N_ROWS * (float)EMBED_DIM));
        out[O_PERP] = expf(-ent);
    }
}

// ---------------------------------------------------------------------------
// kernel_launch
// ---------------------------------------------------------------------------
extern "C" void kernel_launch(void* const* d_in, const int* in_sizes, int n_in,
                              void* d_out, int out_size, void* d_ws, size_t ws_size,
                              hipStream_t stream) {
    const float* x      = (const float*)d_in[0];  // [64,32,32,64]
    const float* emb    = (const float*)d_in[1];  // [1024,64]
    const float* ema_w  = (const float*)d_in[2];  // [1024,64]
    const float* ema_cs = (const float*)d_in[3];  // [1024]
    float* out = (float*)d_out;

    // Workspace layout (floats): wnorm[1024] | dw[65536] | count[1024] | loss[1]
    float* ws_f     = (float*)d_ws;
    float* wnorm    = ws_f;
    float* ws_dw    = ws_f + 1024;
    float* ws_count = ws_f + 1024 + 65536;
    float* ws_loss  = ws_f + 1024 + 65536 + 1024;

    // 1) Zero the one-hot encodings region (268 MB) and ws accumulators.
    zero_f32<<<2048, 256, 0, stream>>>(out + O_ENC, (size_t)N_ROWS * NUM_EMB);
    zero_f32<<<256, 256, 0, stream>>>(ws_dw, (size_t)(65536 + 1024 + 1));

    // 2) Codebook norms.
    wnorm_kernel<<<NUM_EMB / 256, 256, 0, stream>>>(emb, wnorm);

    // 3) WMMA distance GEMM + argmin + gather/scatter.
    vq_main<<<N_BLOCKS, 256, 0, stream>>>(x, emb, wnorm,
                                          out + O_QUANT, out + O_ENC,
                                          ws_count, ws_dw, ws_loss);

    // 4) EMA finalize + scalars.
    vq_finalize<<<1, 1024, 0, stream>>>(ema_w, ema_cs, ws_count, ws_dw, ws_loss, out);
}